// DCTExtractor_19756849561956
// MI455X (gfx1250) — compile-verified
//
#include <hip/hip_runtime.h>

typedef __attribute__((ext_vector_type(2))) float v2f;
typedef __attribute__((ext_vector_type(8))) float v8f;

#define WAVES_PER_BLOCK 8
// 64 images * (512/16)^2 tiles = 65536 tiles, one wave each
#define TOTAL_TILES 65536

__global__ __launch_bounds__(32 * WAVES_PER_BLOCK)
void dct8x8_wmma_f32(const float* __restrict__ x,
                     const float* __restrict__ dctm,   // 8x8 row-major
                     const float* __restrict__ mask,   // 8x8 row-major
                     float* __restrict__ out)
{
    const int lane = threadIdx.x & 31;
    const int wave = threadIdx.x >> 5;
    const int t    = blockIdx.x * WAVES_PER_BLOCK + wave;
    const int img  = t >> 10;          // 64 images
    const int ty   = (t >> 5) & 31;    // 32 tile rows
    const int tx   = t & 31;           // 32 tile cols

    const int h  = lane >> 4;          // half-wave id (0/1)
    const int m  = lane & 15;          // A-layout row / C-D layout column
    const int l7 = lane & 7;

    // ---- constant fragments of block-diag DCT matrix Ď = diag(D, D) ----
    // Needed element (for both products): Ď[lane&15][4b + v + 2h]
    // D columns touched per lane: 2h+{0,1,4,5}; row l&7. Gate by block match.
    const float dv0 = dctm[l7 * 8 + 2 * h + 0];
    const float dv1 = dctm[l7 * 8 + 2 * h + 1];
    const float dv2 = dctm[l7 * 8 + 2 * h + 4];
    const float dv3 = dctm[l7 * 8 + 2 * h + 5];
    const int blkrow = (lane >> 3) & 1;           // (lane&15)>>3
    const float g0 = (blkrow == 0) ? 1.0f : 0.0f; // K-slab 0..7  valid
    const float g1 = (blkrow == 1) ? 1.0f : 0.0f; // K-slab 8..15 valid
    v2f dfrag[4];
    dfrag[0] = v2f{dv0 * g0, dv1 * g0};
    dfrag[1] = v2f{dv2 * g0, dv3 * g0};
    dfrag[2] = v2f{dv0 * g1, dv1 * g1};
    dfrag[3] = v2f{dv2 * g1, dv3 * g1};

    // ---- mask column for this lane (C/D layout: element (r+8h, m), &7 -> (r, l&7)) ----
    float mk[8];
#pragma unroll
    for (int r = 0; r < 8; ++r) mk[r] = mask[r * 8 + l7];

    // ---- grayscale fragments of G in A-layout: G[m][4b + v + 2h] ----
    const long imgBase = (long)img * 3 * 262144;
    const int  y       = ty * 16 + m;
    const int  xb      = tx * 16 + 2 * h;
    const float* p0 = x + imgBase + 0 * 262144 + y * 512 + xb;
    const float* p1 = p0 + 262144;
    const float* p2 = p1 + 262144;
    v2f ga[4];
#pragma unroll
    for (int b = 0; b < 4; ++b) {
        v2f cr = __builtin_nontemporal_load((const v2f*)(p0 + 4 * b));
        v2f cg = __builtin_nontemporal_load((const v2f*)(p1 + 4 * b));
        v2f cb = __builtin_nontemporal_load((const v2f*)(p2 + 4 * b));
        ga[b].x = 0.299f * cr.x + 0.587f * cg.x + 0.114f * cb.x;
        ga[b].y = 0.299f * cr.y + 0.587f * cg.y + 0.114f * cb.y;
    }

    // ---- product 1: W1 = G * Ď^T  (A = G slabs, B = Ď^T slabs == dfrag) ----
    v8f w = {0.f, 0.f, 0.f, 0.f, 0.f, 0.f, 0.f, 0.f};
#pragma unroll
    for (int b = 0; b < 4; ++b)
        w = __builtin_amdgcn_wmma_f32_16x16x4_f32(false, ga[b], false, dfrag[b],
                                                  (short)0, w, false, false);

    // ---- re-layout W1: C/D layout -> B-layout slabs via half-wave exchange ----
    float ws[8];
#pragma unroll
    for (int r = 0; r < 8; ++r) ws[r] = __shfl_xor(w[r], 16, 32);

    v2f bw[4];
#pragma unroll
    for (int b = 0; b < 4; ++b) {
#pragma unroll
        for (int v = 0; v < 2; ++v) {
            const int r0 = (4 * b + v) & 7;      // row wanted by h==0 lanes (mod 8)
            const int r1 = (4 * b + v + 2) & 7;  // row wanted by h==1 lanes (mod 8)
            const float lo = (b < 2) ? w[r0] : ws[r0];
            const float hi = (b < 2) ? ws[r1] : w[r1];
            bw[b][v] = h ? hi : lo;
        }
    }

    // ---- product 2: Y = Ď * W1  (A = Ď slabs == dfrag, B = W1 slabs) ----
    v8f yv = {0.f, 0.f, 0.f, 0.f, 0.f, 0.f, 0.f, 0.f};
#pragma unroll
    for (int b = 0; b < 4; ++b)
        yv = __builtin_amdgcn_wmma_f32_16x16x4_f32(false, dfrag[b], false, bw[b],
                                                   (short)0, yv, false, false);

    // ---- mask + store: lane holds Y[r + 8h][m] in component r ----
    float* op = out + (long)img * 262144 + (long)(ty * 16 + 8 * h) * 512 + tx * 16 + m;
#pragma unroll
    for (int r = 0; r < 8; ++r)
        __builtin_nontemporal_store(yv[r] * mk[r], op + r * 512);
}

extern "C" void kernel_launch(void* const* d_in, const int* in_sizes, int n_in,
                              void* d_out, int out_size, void* d_ws, size_t ws_size,
                              hipStream_t stream) {
    const float* x    = (const float*)d_in[0];
    const float* dctm = (const float*)d_in[1];
    const float* mask = (const float*)d_in[2];
    float* out        = (float*)d_out;

    dim3 grid(TOTAL_TILES / WAVES_PER_BLOCK);
    dim3 block(32 * WAVES_PER_BLOCK);
    dct8x8_wmma_f32<<<grid, block, 0, stream>>>(x, dctm, mask, out);
}